// ScaledDotProductAttention_65386582114570
// MI455X (gfx1250) — compile-verified
//
#include <hip/hip_runtime.h>

typedef __attribute__((ext_vector_type(2))) float v2f;
typedef __attribute__((ext_vector_type(8))) float v8f;

#define NB 32
#define SQ 2048
#define SKK 2048
#define DD 64
#define WAVES 8
#define COLS_PER_WAVE (SKK / WAVES)   // 256
#define NTILES (COLS_PER_WAVE / 16)   // 16

__global__ __launch_bounds__(256, 1) void attn_softmax_kernel(
    const float* __restrict__ q, const float* __restrict__ kmat,
    const int* __restrict__ mask, float* __restrict__ out)
{
    __shared__ float red_max[WAVES][16];
    __shared__ float red_sum[WAVES][16];

    const int qt   = blockIdx.x;      // q-tile index, 0..127
    const int b    = blockIdx.y;      // batch, 0..31
    const int tid  = threadIdx.x;
    const int wave = tid >> 5;
    const int lane = tid & 31;
    const int half = lane >> 4;       // 0 or 1
    const int l16  = lane & 15;

    const int qbase  = qt * 16;
    const int skbase = wave * COLS_PER_WAVE;

    // ---- preload 16x64 Q tile into WMMA-A layout registers ----
    // A-matrix 16x4 f32: lane holds row M=l16; VGPR0 -> K = half*2, VGPR1 -> K = half*2+1
    const float* qptr = q + ((size_t)b * SQ + (size_t)(qbase + l16)) * DD + half * 2;
    v2f aq[16];
#pragma unroll
    for (int kk = 0; kk < 16; ++kk)
        aq[kk] = *(const v2f*)(qptr + 4 * kk);

    // ---- QK^T: 16 tiles of 16x16, K-dim (D=64) in steps of 4 ----
    v8f acc[NTILES];
    const float* kp0 = kmat + (size_t)b * SKK * DD + half * 2;
#pragma unroll
    for (int t = 0; t < NTILES; ++t) {
        const float* kptr = kp0 + (size_t)(skbase + t * 16 + l16) * DD;
        v8f c = {};
#pragma unroll
        for (int kk = 0; kk < 16; ++kk) {
            v2f bb = *(const v2f*)(kptr + 4 * kk);
            c = __builtin_amdgcn_wmma_f32_16x16x4_f32(
                    /*neg_a=*/false, aq[kk], /*neg_b=*/false, bb,
                    /*c_mod=*/(short)0, c, /*reuse_a=*/false, /*reuse_b=*/false);
        }
        acc[t] = c;
    }

    // ---- scale + mask + per-row partial max ----
    // C/D layout: VGPR r of lane L -> element (M = r + (L/16)*8, N = L%16)
    const float scale = 0.125f;  // 1/sqrt(64)
    const float NEGINF = -__builtin_inff();
    float rmax[8];
#pragma unroll
    for (int r = 0; r < 8; ++r) rmax[r] = NEGINF;

    const size_t mrow0 = ((size_t)b * SQ + (size_t)qbase) * (size_t)SKK;
#pragma unroll
    for (int t = 0; t < NTILES; ++t) {
        const int col = skbase + t * 16 + l16;
#pragma unroll
        for (int r = 0; r < 8; ++r) {
            const int row = r + half * 8;
            // streaming read-once data: non-temporal, keep L2 for K
            const int m = __builtin_nontemporal_load(
                              &mask[mrow0 + (size_t)row * SKK + col]);
            float s = acc[t][r] * scale;
            s = (m == 0) ? NEGINF : s;
            acc[t][r] = s;
            rmax[r] = fmaxf(rmax[r], s);
        }
    }

    // max across the 16 lanes that hold the same row (xor masks stay in-half)
#pragma unroll
    for (int r = 0; r < 8; ++r) {
        float v = rmax[r];
#pragma unroll
        for (int off = 1; off <= 8; off <<= 1)
            v = fmaxf(v, __shfl_xor(v, off, 32));
        rmax[r] = v;
    }
    if (l16 == 0) {
#pragma unroll
        for (int r = 0; r < 8; ++r) red_max[wave][r + half * 8] = rmax[r];
    }
    __syncthreads();

    float gmax[8];
#pragma unroll
    for (int r = 0; r < 8; ++r) {
        float v = NEGINF;
#pragma unroll
        for (int w = 0; w < WAVES; ++w) v = fmaxf(v, red_max[w][r + half * 8]);
        gmax[r] = v;
    }

    // ---- exp + per-row partial sum ----
    const float LOG2E = 1.4426950408889634f;
    float rsum[8];
#pragma unroll
    for (int r = 0; r < 8; ++r) rsum[r] = 0.0f;
#pragma unroll
    for (int t = 0; t < NTILES; ++t) {
#pragma unroll
        for (int r = 0; r < 8; ++r) {
            float p = __builtin_amdgcn_exp2f((acc[t][r] - gmax[r]) * LOG2E);
            acc[t][r] = p;
            rsum[r] += p;
        }
    }
#pragma unroll
    for (int r = 0; r < 8; ++r) {
        float v = rsum[r];
#pragma unroll
        for (int off = 1; off <= 8; off <<= 1)
            v += __shfl_xor(v, off, 32);
        rsum[r] = v;
    }
    if (l16 == 0) {
#pragma unroll
        for (int r = 0; r < 8; ++r) red_sum[wave][r + half * 8] = rsum[r];
    }
    __syncthreads();

    float rinv[8];
#pragma unroll
    for (int r = 0; r < 8; ++r) {
        float v = 0.0f;
#pragma unroll
        for (int w = 0; w < WAVES; ++w) v += red_sum[w][r + half * 8];
        rinv[r] = __builtin_amdgcn_rcpf(v);
    }

    // ---- normalize + store (non-temporal, half-wave coalesced 64B segments) ----
#pragma unroll
    for (int t = 0; t < NTILES; ++t) {
        const int col = skbase + t * 16 + l16;
#pragma unroll
        for (int r = 0; r < 8; ++r) {
            const int row = r + half * 8;
            __builtin_nontemporal_store(acc[t][r] * rinv[r],
                &out[mrow0 + (size_t)row * SKK + col]);
        }
    }
}

extern "C" void kernel_launch(void* const* d_in, const int* in_sizes, int n_in,
                              void* d_out, int out_size, void* d_ws, size_t ws_size,
                              hipStream_t stream) {
    const float* q    = (const float*)d_in[0];
    const float* k    = (const float*)d_in[1];
    const int*   mask = (const int*)d_in[2];
    float*       out  = (float*)d_out;

    dim3 grid(SQ / 16, NB, 1);
    dim3 block(256, 1, 1);
    attn_softmax_kernel<<<grid, block, 0, stream>>>(q, k, mask, out);
}